// GraphNeuralNetwork_33964601377430
// MI455X (gfx1250) — compile-verified
//
#include <hip/hip_runtime.h>
#include <hip/hip_bf16.h>

#define NN  100000
#define NE  3200000
#define FIN 128
#define FH  64

typedef __attribute__((ext_vector_type(2))) float v2f;
typedef __attribute__((ext_vector_type(8))) float v8f;

// ---------------- degree / norm ----------------
__global__ void k_deg_init(float* __restrict__ deg) {
  int i = blockIdx.x * blockDim.x + threadIdx.x;
  if (i < NN) deg[i] = 1.0f;               // self-loop contributes 1
}

__global__ void k_deg_acc(const int* __restrict__ ei, float* __restrict__ deg) {
  int e = blockIdx.x * blockDim.x + threadIdx.x;
  if (e < NE) atomicAdd(&deg[ei[NE + e]], 1.0f);   // dst = ei[1][e]
}

__global__ void k_dinv(float* __restrict__ deg) {
  int i = blockIdx.x * blockDim.x + threadIdx.x;
  if (i < NN) deg[i] = rsqrtf(deg[i]);     // deg >= 1 always (self loops)
}

// ---------------- dense GEMM via f32 WMMA ----------------
// H[M x 64] = X[M x K] @ W[64 x K]^T ; M = 100000 = 6250 tiles of 16.
// One workgroup (4 waves) per M-tile; wave w handles N-tile w (cols 16w..16w+15).
template<int K>
__global__ __launch_bounds__(128) void k_gemm_wmma(const float* __restrict__ X,
                                                   const float* __restrict__ W,
                                                   float* __restrict__ H) {
  const int lane = threadIdx.x & 31;
  const int wave = threadIdx.x >> 5;        // 0..3 -> N tile
  const int m16  = lane & 15;
  const int koff = (lane >> 4) << 1;        // lanes 16-31 hold K+2,K+3
  const float* xr = X + (size_t)(blockIdx.x * 16 + m16) * K;  // A row
  const float* wr = W + (size_t)(wave * 16 + m16) * K;        // B col (= W row)
  v8f c = {0.f, 0.f, 0.f, 0.f, 0.f, 0.f, 0.f, 0.f};
#pragma unroll
  for (int k = 0; k < K; k += 4) {
    v2f a; a.x = xr[k + koff]; a.y = xr[k + koff + 1];
    v2f b; b.x = wr[k + koff]; b.y = wr[k + koff + 1];
    // D = A(16x4,f32) * B(4x16,f32) + C
    c = __builtin_amdgcn_wmma_f32_16x16x4_f32(false, a, false, b,
                                              (short)0, c, false, false);
  }
  // D layout: VGPR v holds row (v + 8*(lane>>4)) of the tile, col = lane&15
  float* hp = H + (size_t)blockIdx.x * 16 * FH + wave * 16 + m16;
  const int r0 = (lane >> 4) * 8;
#pragma unroll
  for (int v = 0; v < 8; ++v)
    hp[(size_t)(r0 + v) * FH] = c[v];
}

// ---------------- propagate: self-loop init + edge scatter ----------------
// agg[i,:] = h[i,:] * dinv[i]^2   (self-loop term), float4 per thread
__global__ void k_agg_init(const float* __restrict__ h, const float* __restrict__ dinv,
                           float* __restrict__ agg) {
  int i = blockIdx.x * blockDim.x + threadIdx.x;   // over NN*16 float4 slots
  if (i >= NN * 16) return;
  const int node = i >> 4;
  float w = dinv[node]; w *= w;
  float4 hv = ((const float4*)h)[i];
  float4 o;
  o.x = hv.x * w; o.y = hv.y * w; o.z = hv.z * w; o.w = hv.w * w;
  ((float4*)agg)[i] = o;
}

// 16 threads per edge, float4 each: gather h[src], scale, atomic-add into agg[dst]
__global__ void k_scatter(const int* __restrict__ ei, const float* __restrict__ dinv,
                          const float* __restrict__ h, float* __restrict__ agg) {
  long long gid = (long long)blockIdx.x * blockDim.x + threadIdx.x;
  if (gid >= (long long)NE * 16) return;
  const int e  = (int)(gid >> 4);
  const int f4 = ((int)gid & 15) * 4;
  const int s = ei[e];
  const int d = ei[NE + e];
  const float w = dinv[s] * dinv[d];
  float4 hv = *(const float4*)(h + (size_t)s * FH + f4);
  float* ap = agg + (size_t)d * FH + f4;
  atomicAdd(ap + 0, hv.x * w);
  atomicAdd(ap + 1, hv.y * w);
  atomicAdd(ap + 2, hv.z * w);
  atomicAdd(ap + 3, hv.w * w);
}

// o = relu(agg + b)
__global__ void k_bias_relu(const float* __restrict__ agg, const float* __restrict__ b,
                            float* __restrict__ o) {
  int i = blockIdx.x * blockDim.x + threadIdx.x;
  if (i >= NN * FH) return;
  float v = agg[i] + b[i & (FH - 1)];
  o[i] = v > 0.f ? v : 0.f;
}

// out[i] = (agg[i,:] + b2) . Wc + bc
__global__ void k_head(const float* __restrict__ agg, const float* __restrict__ b2,
                       const float* __restrict__ Wc, const float* __restrict__ bc,
                       float* __restrict__ out) {
  int i = blockIdx.x * blockDim.x + threadIdx.x;
  if (i >= NN) return;
  const float4* a = (const float4*)(agg + (size_t)i * FH);
  float s = bc[0];
#pragma unroll
  for (int f = 0; f < FH / 4; ++f) {
    float4 av = a[f];
    float4 bv = ((const float4*)b2)[f];
    float4 wv = ((const float4*)Wc)[f];
    s += (av.x + bv.x) * wv.x + (av.y + bv.y) * wv.y +
         (av.z + bv.z) * wv.z + (av.w + bv.w) * wv.w;
  }
  out[i] = s;
}

extern "C" void kernel_launch(void* const* d_in, const int* in_sizes, int n_in,
                              void* d_out, int out_size, void* d_ws, size_t ws_size,
                              hipStream_t stream) {
  const float* x  = (const float*)d_in[0];
  const int*   ei = (const int*)d_in[1];    // [2, NE] int32 (jax default x64-off)
  const float* W1 = (const float*)d_in[2];
  const float* b1 = (const float*)d_in[3];
  const float* W2 = (const float*)d_in[4];
  const float* b2 = (const float*)d_in[5];
  const float* Wc = (const float*)d_in[6];
  const float* bc = (const float*)d_in[7];
  float* out = (float*)d_out;

  char* ws = (char*)d_ws;
  float* dinv = (float*)ws;                                      // NN floats
  float* A = (float*)(ws + (1u << 20));                          // NN*FH floats
  float* B = (float*)(ws + (1u << 20) + (size_t)NN * FH * 4);    // NN*FH floats

  const int TB = 256;
  // ---- normalization coefficients ----
  k_deg_init<<<(NN + TB - 1) / TB, TB, 0, stream>>>(dinv);
  k_deg_acc <<<(NE + TB - 1) / TB, TB, 0, stream>>>(ei, dinv);
  k_dinv    <<<(NN + TB - 1) / TB, TB, 0, stream>>>(dinv);

  const int ETHREADS_BLOCKS = (int)(((long long)NE * 16 + TB - 1) / TB);

  // ---- layer 1: A = x @ W1^T ; B = propagate(A) ; A = relu(B + b1) ----
  k_gemm_wmma<FIN><<<NN / 16, 128, 0, stream>>>(x, W1, A);
  k_agg_init <<<(NN * 16 + TB - 1) / TB, TB, 0, stream>>>(A, dinv, B);
  k_scatter  <<<ETHREADS_BLOCKS, TB, 0, stream>>>(ei, dinv, A, B);
  k_bias_relu<<<(NN * FH + TB - 1) / TB, TB, 0, stream>>>(B, b1, A);

  // ---- layer 2: B = A @ W2^T ; A = propagate(B) ----
  k_gemm_wmma<FH><<<NN / 16, 128, 0, stream>>>(A, W2, B);
  k_agg_init <<<(NN * 16 + TB - 1) / TB, TB, 0, stream>>>(B, dinv, A);
  k_scatter  <<<ETHREADS_BLOCKS, TB, 0, stream>>>(ei, dinv, B, A);

  // ---- head: out = (A + b2) @ Wc^T + bc ----
  k_head<<<(NN + TB - 1) / TB, TB, 0, stream>>>(A, b2, Wc, bc, out);
}